// NERModelWithCRF_67937792688590
// MI455X (gfx1250) — compile-verified
//
#include <hip/hip_runtime.h>
#include <hip/hip_bf16.h>

typedef float v2f __attribute__((ext_vector_type(2)));
typedef float v8f __attribute__((ext_vector_type(8)));
typedef unsigned int v4u __attribute__((ext_vector_type(4)));
typedef int v4i __attribute__((ext_vector_type(4)));
typedef int v8i __attribute__((ext_vector_type(8)));

#define B_ 64
#define T_ 512
#define H_ 768
#define L_ 9

#if defined(__gfx1250__) && __has_builtin(__builtin_amdgcn_tensor_load_to_lds)
#define USE_TDM 1
#endif

#ifdef USE_TDM
// ---------------------------------------------------------------------------
// Issue a TDM load of a 16-row x 32-col f32 tile of `hidden` (row stride H_)
// into LDS at byte address lds_byte_addr, with hardware padding of 1 DWORD
// every 32 DWORDs (=> LDS row stride 33 floats, bank-conflict free).
// Descriptor built per CDNA5 ISA; all words wave-uniform,
// forced to SGPRs via readfirstlane. 6-arg builtin form (clang-23 / therock).
// ---------------------------------------------------------------------------
__device__ __forceinline__ void tdm_issue_tile(const float* gsrc, unsigned lds_byte_addr)
{
    const unsigned long long ga = (unsigned long long)(const void*)gsrc;
    v4u g0;
    g0[0] = 1u;                                                   // count=1, user mode
    g0[1] = (unsigned)__builtin_amdgcn_readfirstlane((int)lds_byte_addr);
    g0[2] = (unsigned)__builtin_amdgcn_readfirstlane((int)(unsigned)ga);
    g0[3] = ((unsigned)__builtin_amdgcn_readfirstlane((int)(unsigned)(ga >> 32)) & 0x1FFFFFFu)
            | 0x80000000u;                                        // type=2 ("image")
    v8i g1;
    g1[0] = 0x01120000;        // data_size=4B | pad_enable | pad_interval=32dw (pad_amount=1dw)
    g1[1] = (int)(768u << 16); // tensor_dim0 = H_ (bits 79:48, low half)
    g1[2] = (int)0x80000000u;  // tensor_dim1 = 32768 rows (bits 95:80)
    g1[3] = (int)(32u << 16);  // tile_dim0 = 32 (bits 127:112)
    g1[4] = 16;                // tile_dim1 = 16 rows (bits 143:128)
    g1[5] = H_;                // tensor_dim0_stride = 768 (bits 207:160, low 32)
    g1[6] = 0;
    g1[7] = 0;
    const v4i z4 = {0, 0, 0, 0};
    const v8i z8 = {0, 0, 0, 0, 0, 0, 0, 0};
    asm volatile("" ::: "memory");
    __builtin_amdgcn_tensor_load_to_lds(g0, g1, z4, z4, z8, 0);
    asm volatile("" ::: "memory");
}
#endif

// ---------------------------------------------------------------------------
// Kernel 1: logits[b,t,:] = hidden[b,t,:] @ W^T + bias via V_WMMA_F32_16X16X4_F32.
// One wave per 16(M) x 16(N) tile (9 labels zero-padded to 16); K=768 in 32-wide
// chunks (8 WMMAs each). Hidden tiles reach LDS via TDM (double-buffered) or a
// manual coalesced stage; W^T is zero-padded into LDS once per block so B-frag
// reads are branchless, conflict-free ds_load_2addr.
// ---------------------------------------------------------------------------
__global__ __launch_bounds__(256) void logits_wmma_kernel(
    const float* __restrict__ hidden, const float* __restrict__ W,
    const float* __restrict__ bias, float* __restrict__ logits)
{
    __shared__ float Wlds[H_ * 16];         // 48 KB: Wt[k][n], n>=9 zeroed
    __shared__ float tile[8][2][16 * 33];   // 33 KB: per-wave double-buffered A tile

    const int wave = threadIdx.x >> 5;
    const int lane = threadIdx.x & 31;
    const int row0 = (blockIdx.x * 8 + wave) * 16;    // 2048 tiles over 256 blocks
    const int n = lane & 15;
    const int khalf = (lane >> 4) * 2;                // lanes 16-31 hold K+2,K+3

    // Stage padded W^T (branchless: clamped address + cndmask, no exec branches).
    for (int idx = threadIdx.x; idx < H_ * 16; idx += 256) {
        const int k = idx >> 4, nn = idx & 15;
        const float wv = W[(size_t)(nn < L_ ? nn : 0) * H_ + k];
        Wlds[idx] = (nn < L_) ? wv : 0.0f;
    }
    __syncthreads();

    const float bv = bias[n < L_ ? n : 0];
    const float bn = (n < L_) ? bv : 0.0f;
    v8f acc;
#pragma unroll
    for (int v = 0; v < 8; ++v) acc[v] = bn;

    // 8 x K=4 WMMA steps over one staged 32-wide chunk.
    auto compute_chunk = [&](const float* my, int k0) {
#pragma unroll
        for (int ks = 0; ks < 8; ++ks) {
            const int k = ks * 4 + khalf;
            v2f a;                                      // A frag: 16x4 f32
            a.x = my[(lane & 15) * 33 + k];
            a.y = my[(lane & 15) * 33 + k + 1];
            const int kg = k0 + k;
            v2f bf;                                     // B frag: 4x16 f32
            bf.x = Wlds[kg * 16 + n];
            bf.y = Wlds[(kg + 1) * 16 + n];
            acc = __builtin_amdgcn_wmma_f32_16x16x4_f32(
                false, a, false, bf, (short)0, acc, false, false);
        }
    };

#ifdef USE_TDM
    // Double-buffered TDM pipeline: DMA chunk k0+32 while WMMAs consume chunk k0.
    const unsigned lb0 = (unsigned)(size_t)&tile[wave][0][0];
    const unsigned lb1 = (unsigned)(size_t)&tile[wave][1][0];
    const float* gbase = hidden + (size_t)row0 * H_;
    tdm_issue_tile(gbase, lb0);
    int buf = 0;
    for (int k0 = 0; k0 < H_; k0 += 32) {
        if (k0 + 32 < H_) {
            tdm_issue_tile(gbase + (k0 + 32), buf ? lb0 : lb1);
            __builtin_amdgcn_s_wait_tensorcnt(1);   // oldest (current buf) complete
        } else {
            __builtin_amdgcn_s_wait_tensorcnt(0);
        }
        compute_chunk(tile[wave][buf], k0);
        buf ^= 1;
    }
#else
    // Fallback: manual coalesced stage (128B per row-load) into buffer 0.
    float* my = tile[wave][0];
    for (int k0 = 0; k0 < H_; k0 += 32) {
#pragma unroll
        for (int r = 0; r < 16; ++r)
            my[r * 33 + lane] = hidden[(size_t)(row0 + r) * H_ + k0 + lane];
        asm volatile("s_wait_dscnt 0" ::: "memory");
        compute_chunk(my, k0);
    }
#endif

    // D layout: lane l, VGPR v -> M = (l<16 ? v : v+8), N = l&15
#pragma unroll
    for (int v = 0; v < 8; ++v) {
        const int m = (lane < 16) ? v : v + 8;
        logits[(size_t)(row0 + m) * 16 + n] = acc[v];
    }
}

// ---------------------------------------------------------------------------
// Kernel 2: CRF NLL. One wave per batch row; lane j = tag j (j<9 active).
// ---------------------------------------------------------------------------
__global__ __launch_bounds__(32) void crf_nll_kernel(
    const float* __restrict__ logits, const float* __restrict__ trans,
    const int* __restrict__ labels, const int* __restrict__ attn,
    float* __restrict__ loss_out)
{
    const int b = blockIdx.x;
    const int lane = threadIdx.x;
    const int j = lane;
    const bool active = j < L_;

    float tcol[L_];                              // transitions[:, j]
#pragma unroll
    for (int i = 0; i < L_; ++i) tcol[i] = trans[i * L_ + (active ? j : 0)];

    const float* lg = logits + (size_t)b * T_ * 16;
    const int* lab  = labels + b * T_;
    const int* am   = attn + b * T_;

    // ---- numerator: emit_score + trans_score (lane-parallel over t) ----
    float num = 0.0f;
    for (int t = lane; t < T_; t += 32) {
        const int l = lab[t];
        const bool mk = (am[t] != 0) && (l != -100);
        if (mk) {
            const int tag = (l == -100) ? 0 : l;
            num += lg[t * 16 + tag];
            if (t >= 1) {
                const int lp = lab[t - 1];
                const int tagp = (lp == -100) ? 0 : lp;
                num += trans[tagp * L_ + tag];
            }
        }
    }
#pragma unroll
    for (int off = 16; off > 0; off >>= 1) num += __shfl_xor(num, off, 32);

    // ---- alpha scan (stabilized logsumexp over 9 states) ----
    float alpha = active ? lg[j] : -1e30f;
    for (int t = 1; t < T_; ++t) {
        const int l = lab[t];
        const bool mk = (am[t] != 0) && (l != -100);
        float av[L_];
#pragma unroll
        for (int i = 0; i < L_; ++i) av[i] = __shfl(alpha, i, 32);
        float m = -1e30f;
#pragma unroll
        for (int i = 0; i < L_; ++i) { const float x = av[i] + tcol[i]; av[i] = x; m = fmaxf(m, x); }
        float s = 0.0f;
#pragma unroll
        for (int i = 0; i < L_; ++i) s += __expf(av[i] - m);
        const float na = m + __logf(s) + (active ? lg[t * 16 + j] : 0.0f);
        if (mk && active) alpha = na;
    }

    // denominator = logsumexp(alpha)
    float m = -1e30f;
    float av[L_];
#pragma unroll
    for (int i = 0; i < L_; ++i) { av[i] = __shfl(alpha, i, 32); m = fmaxf(m, av[i]); }
    float s = 0.0f;
#pragma unroll
    for (int i = 0; i < L_; ++i) s += __expf(av[i] - m);
    const float denom = m + __logf(s);

    if (lane == 0) atomicAdd(loss_out, -(num - denom));
}

// ---------------------------------------------------------------------------
// Kernel 3: Viterbi + backtrace + shifted predictions. One wave per batch.
// Backpointers kept in LDS so the serial backtrace never leaves the WGP.
// ---------------------------------------------------------------------------
__global__ __launch_bounds__(32) void viterbi_kernel(
    const float* __restrict__ logits, const float* __restrict__ trans,
    const int* __restrict__ labels, const int* __restrict__ attn,
    float* __restrict__ pred)     // pred[b*T + t], float-encoded ints
{
    __shared__ unsigned char bp[(T_ - 1) * L_];   // 4599 B
    __shared__ unsigned char mks[T_];

    const int b = blockIdx.x;
    const int lane = threadIdx.x;
    const int j = lane;
    const bool active = j < L_;

    float tcol[L_];
#pragma unroll
    for (int i = 0; i < L_; ++i) tcol[i] = trans[i * L_ + (active ? j : 0)];

    const float* lg = logits + (size_t)b * T_ * 16;
    const int* lab  = labels + b * T_;
    const int* am   = attn + b * T_;

    for (int t = lane; t < T_; t += 32)
        mks[t] = (unsigned char)((am[t] != 0) && (lab[t] != -100));
    asm volatile("s_wait_dscnt 0" ::: "memory");

    float vs = active ? lg[j] : -1e30f;
    for (int t = 1; t < T_; ++t) {
        const bool mk = mks[t] != 0;
        float av[L_];
#pragma unroll
        for (int i = 0; i < L_; ++i) av[i] = __shfl(vs, i, 32);
        float best = av[0] + tcol[0];
        int bi = 0;
#pragma unroll
        for (int i = 1; i < L_; ++i) {
            const float x = av[i] + tcol[i];
            if (x > best) { best = x; bi = i; }   // strict '>' == first-max (jnp.argmax)
        }
        const float nv = best + (active ? lg[t * 16 + j] : 0.0f);
        if (!mk) bi = j;                          // identity backpointer when masked
        else if (active) vs = nv;
        if (active) bp[(t - 1) * L_ + j] = (unsigned char)bi;
    }
    asm volatile("s_wait_dscnt 0" ::: "memory");

    // last_tag = argmax_j vscore (first max)
    float bestv = -1e30f;
    int last = 0;
#pragma unroll
    for (int i = 0; i < L_; ++i) {
        const float x = __shfl(vs, i, 32);
        if (x > bestv) { bestv = x; last = i; }
    }

    if (lane == 0) {
        int tag = last;                           // tag at position T-1 (unused in preds)
        pred[(size_t)b * T_ + 0] = -100.0f;
        for (int r = T_ - 2; r >= 0; --r) {
            tag = bp[r * L_ + tag];               // tag at position r
            pred[(size_t)b * T_ + r + 1] = mks[r] ? (float)tag : -100.0f;
        }
    }
}

// ---------------------------------------------------------------------------
extern "C" void kernel_launch(void* const* d_in, const int* in_sizes, int n_in,
                              void* d_out, int out_size, void* d_ws, size_t ws_size,
                              hipStream_t stream)
{
    const float* hidden = (const float*)d_in[0];   // (B,T,H) f32
    const float* W      = (const float*)d_in[1];   // (L,H)   f32
    const float* bias   = (const float*)d_in[2];   // (L,)    f32
    const float* trans  = (const float*)d_in[3];   // (L,L)   f32
    const int*   labels = (const int*)d_in[4];     // (B,T)   i32
    const int*   attn   = (const int*)d_in[5];     // (B,T)   i32

    float* out    = (float*)d_out;                 // [0]=loss, [1..]=predictions
    float* logits = (float*)d_ws;                  // (B*T, 16) padded, 2 MB

    (void)hipMemsetAsync(out, 0, sizeof(float), stream); // zero loss accumulator

    logits_wmma_kernel<<<256, 256, 0, stream>>>(hidden, W, bias, logits);
    crf_nll_kernel<<<B_, 32, 0, stream>>>(logits, trans, labels, attn, out);
    viterbi_kernel<<<B_, 32, 0, stream>>>(logits, trans, labels, attn, out + 1);
}